// LazyMLPBlock_48009144434822
// MI455X (gfx1250) — compile-verified
//
#include <hip/hip_runtime.h>

// MoE block (gpt-oss style): S=32,H=768,I=768,E=32,K=4, fp32.
// Memory-bound (~2 FLOP/byte, >=226MB expert weights): expert-centric tiling,
// each expert's weights streamed once, multiplied vs a 16-token WMMA tile.

typedef __attribute__((ext_vector_type(2))) float v2f;
typedef __attribute__((ext_vector_type(8))) float v8f;

static constexpr int kS = 32;
static constexpr int kH = 768;
static constexpr int kI = 768;
static constexpr int kE = 32;
static constexpr int kK = 4;
static constexpr float kLimit = 7.0f;
static constexpr float kAlpha = 1.702f;
static constexpr float kEps = 1e-5f;

__device__ __forceinline__ float wave_reduce_sum(float v) {
  #pragma unroll
  for (int off = 16; off > 0; off >>= 1) v += __shfl_xor(v, off, 32);
  return v;
}

// ---------------------------------------------------------------------------
// Kernel 1: residual copy + RMSNorm + router (top-4, softmax) + expert lists.
// One block, 256 threads (8 wave32).
// ---------------------------------------------------------------------------
__global__ void moe_router_kernel(const float* __restrict__ x,
                                  const float* __restrict__ norm_scale,
                                  const float* __restrict__ gate_w,
                                  const float* __restrict__ gate_b,
                                  float* __restrict__ out,
                                  float* __restrict__ t_ws,
                                  float* __restrict__ wts,
                                  int* __restrict__ cntArr,
                                  int* __restrict__ tokArr,
                                  int* __restrict__ slotArr) {
  __shared__ float gl[kS * kE];
  __shared__ int cnt[kE];
  const int tid  = threadIdx.x;
  const int lane = tid & 31;
  const int wave = tid >> 5;

  // Residual: out starts as x (ffn2 atomically accumulates on top).
  for (int i = tid; i < kS * kH; i += 256) out[i] = x[i];

  // RMSNorm: wave w handles tokens 4w..4w+3.
  for (int j = 0; j < 4; ++j) {
    const int s = wave * 4 + j;
    const float* xr = x + s * kH;
    float ss = 0.f;
    for (int i = lane; i < kH; i += 32) { float v = xr[i]; ss += v * v; }
    ss = wave_reduce_sum(ss);
    const float rs = rsqrtf(ss / (float)kH + kEps);
    float* tr = t_ws + s * kH;
    for (int i = lane; i < kH; i += 32) tr[i] = xr[i] * rs * norm_scale[i];
  }
  if (tid < kE) cnt[tid] = 0;
  __syncthreads();

  // Gate logits g[s][e] (tiny: 32x32 dots of length 768).
  for (int p = tid; p < kS * kE; p += 256) {
    const int s = p >> 5, e = p & 31;
    const float* tr = t_ws + s * kH;
    const float* gw = gate_w + e * kH;
    float acc = 0.f;
    for (int i = 0; i < kH; ++i) acc += tr[i] * gw[i];
    gl[s * kE + e] = acc + gate_b[e];
  }
  __syncthreads();

  // Top-4 + softmax per token; build per-expert compacted lists.
  if (tid < kS) {
    const int s = tid;
    float v[kE];
    for (int e = 0; e < kE; ++e) v[e] = gl[s * kE + e];
    float tv[kK]; int ti[kK];
    for (int k = 0; k < kK; ++k) {
      float best = -1e30f; int bi = 0;
      for (int e = 0; e < kE; ++e) if (v[e] > best) { best = v[e]; bi = e; }
      tv[k] = best; ti[k] = bi; v[bi] = -1e30f;
    }
    const float m = tv[0];
    float ex[kK], sum = 0.f;
    for (int k = 0; k < kK; ++k) { ex[k] = __expf(tv[k] - m); sum += ex[k]; }
    for (int k = 0; k < kK; ++k) {
      const int e = ti[k];
      const int pos = atomicAdd(&cnt[e], 1);
      tokArr[e * kS + pos]  = s;             // token id
      slotArr[e * kS + pos] = s * kK + k;    // (token,k) pair slot
      wts[s * kK + k] = ex[k] / sum;
    }
  }
  __syncthreads();
  if (tid < kE) cntArr[tid] = cnt[tid];
}

// ---------------------------------------------------------------------------
// Kernel 2: h = w1[e] @ t (+b1), interleaved clamped SwiGLU -> act_ws.
// grid (E, 2I/32), 64 threads (2 waves); wave owns a 16-row strip.
// D(16 rows x 16 tokens) = sum_k A(16x4) x B(4x16), v_wmma_f32_16x16x4_f32.
// Per ISA layout, lane L holds A/B elements K = 2*(L>>4)+{0,1}: float2 loads.
// ---------------------------------------------------------------------------
__global__ void moe_ffn1_kernel(const float* __restrict__ w1,
                                const float* __restrict__ b1,
                                const float* __restrict__ t_ws,
                                float* __restrict__ act_ws,
                                const int* __restrict__ cntArr,
                                const int* __restrict__ tokArr,
                                const int* __restrict__ slotArr) {
  const int e   = blockIdx.x;
  const int cnt = cntArr[e];
  if (cnt == 0) return;                       // uniform per block
  const int lane = threadIdx.x & 31;
  const int wave = threadIdx.x >> 5;
  const int half = lane >> 4;
  const int l16  = lane & 15;
  const int rowBase = blockIdx.y * 32 + wave * 16;   // in [0, 2I)

  const float* aRow =
      w1 + ((size_t)e * 2 * kI + rowBase + l16) * kH + 2 * half;
  const int nct = (cnt + 15) >> 4;
  for (int ct = 0; ct < nct; ++ct) {
    const int n  = ct * 16 + l16;
    const int nc = n < cnt ? n : cnt - 1;     // clamp: keep EXEC all-ones
    const int tok  = tokArr[e * kS + nc];
    const int slot = slotArr[e * kS + nc];
    const float* bRow = t_ws + tok * kH + 2 * half;

    v8f acc = {};
    #pragma unroll 4
    for (int k = 0; k < kH; k += 4) {
      v2f a = *(const v2f*)(aRow + k);
      v2f b = *(const v2f*)(bRow + k);
      acc = __builtin_amdgcn_wmma_f32_16x16x4_f32(
          false, a, false, b, (short)0, acc, false, false);
    }

    if (n < cnt) {
      // acc[j] is row (rowBase + j + 8*half), column = token n.
      // j even/odd pairs are the interleaved (glu, lin) rows.
      #pragma unroll
      for (int j = 0; j < 8; j += 2) {
        const int row = rowBase + j + 8 * half;      // even row
        float hg = acc[j]     + b1[e * 2 * kI + row];
        float hl = acc[j + 1] + b1[e * 2 * kI + row + 1];
        hg = fminf(hg, kLimit);
        hl = fminf(fmaxf(hl, -kLimit), kLimit);
        const float sg = 1.f / (1.f + __expf(-kAlpha * hg));
        act_ws[slot * kI + (row >> 1)] = hg * sg * (hl + 1.f);
      }
    }
  }
}

// ---------------------------------------------------------------------------
// Kernel 3: o = w2[e] @ act (+b2); out[token] += wt * o (atomic).
// grid (E, H/32), 64 threads (2 waves).
// ---------------------------------------------------------------------------
__global__ void moe_ffn2_kernel(const float* __restrict__ w2,
                                const float* __restrict__ b2,
                                const float* __restrict__ act_ws,
                                const float* __restrict__ wts,
                                const int* __restrict__ cntArr,
                                const int* __restrict__ tokArr,
                                const int* __restrict__ slotArr,
                                float* __restrict__ out) {
  const int e   = blockIdx.x;
  const int cnt = cntArr[e];
  if (cnt == 0) return;
  const int lane = threadIdx.x & 31;
  const int wave = threadIdx.x >> 5;
  const int half = lane >> 4;
  const int l16  = lane & 15;
  const int rowBase = blockIdx.y * 32 + wave * 16;   // in [0, H)

  const float* aRow =
      w2 + ((size_t)e * kH + rowBase + l16) * kI + 2 * half;
  const int nct = (cnt + 15) >> 4;
  for (int ct = 0; ct < nct; ++ct) {
    const int n  = ct * 16 + l16;
    const int nc = n < cnt ? n : cnt - 1;
    const int tok  = tokArr[e * kS + nc];
    const int slot = slotArr[e * kS + nc];
    const float w = wts[slot];
    const float* bRow = act_ws + slot * kI + 2 * half;

    v8f acc = {};
    #pragma unroll 4
    for (int k = 0; k < kI; k += 4) {
      v2f a = *(const v2f*)(aRow + k);
      v2f b = *(const v2f*)(bRow + k);
      acc = __builtin_amdgcn_wmma_f32_16x16x4_f32(
          false, a, false, b, (short)0, acc, false, false);
    }

    if (n < cnt) {
      #pragma unroll
      for (int j = 0; j < 8; ++j) {
        const int row = rowBase + j + 8 * half;
        atomicAdd(out + tok * kH + row, w * (acc[j] + b2[e * kH + row]));
      }
    }
  }
}

// ---------------------------------------------------------------------------
extern "C" void kernel_launch(void* const* d_in, const int* in_sizes, int n_in,
                              void* d_out, int out_size, void* d_ws, size_t ws_size,
                              hipStream_t stream) {
  (void)in_sizes; (void)n_in; (void)out_size; (void)ws_size;
  const float* x          = (const float*)d_in[0];
  const float* norm_scale = (const float*)d_in[1];
  const float* gate_w     = (const float*)d_in[2];
  const float* gate_b     = (const float*)d_in[3];
  const float* w1         = (const float*)d_in[4];
  const float* b1         = (const float*)d_in[5];
  const float* w2         = (const float*)d_in[6];
  const float* b2         = (const float*)d_in[7];
  float* out = (float*)d_out;

  char* ws = (char*)d_ws;
  float* t_ws   = (float*)(ws);            // 32*768*4      = 98304 B
  float* act_ws = (float*)(ws + 98304);    // 128*768*4     = 393216 B
  float* wts    = (float*)(ws + 491520);   // 128*4         = 512 B
  int*   cntArr = (int*)  (ws + 492032);   // 32*4
  int*   tokArr = (int*)  (ws + 492160);   // 32*32*4
  int*   slotArr= (int*)  (ws + 496256);   // 32*32*4  -> total ~500 KB

  moe_router_kernel<<<1, 256, 0, stream>>>(x, norm_scale, gate_w, gate_b, out,
                                           t_ws, wts, cntArr, tokArr, slotArr);
  moe_ffn1_kernel<<<dim3(kE, (2 * kI) / 32), 64, 0, stream>>>(
      w1, b1, t_ws, act_ws, cntArr, tokArr, slotArr);
  moe_ffn2_kernel<<<dim3(kE, kH / 32), 64, 0, stream>>>(
      w2, b2, act_ws, wts, cntArr, tokArr, slotArr, out);
}